// ParaformerV2_52209622450684
// MI455X (gfx1250) — compile-verified
//
#include <hip/hip_runtime.h>
#include <stdint.h>

#define B_ 32
#define T_ 2048
#define V_ 513
#define BLK 256

// ---------------- CDNA5 (gfx1250) async global->LDS helpers ----------------
// GLOBAL_LOAD_ASYNC_TO_LDS_B32: per-lane copy memory->LDS, tracked with
// ASYNCcnt (in-order completion). VDST = LDS byte address VGPR, VADDR = 64-bit
// global address ("off" form). b32 chosen because rows are 2052 B (4 B aligned).
__device__ __forceinline__ void async_ld_b32(uint32_t lds_off, const float* gaddr) {
  asm volatile("global_load_async_to_lds_b32 %0, %1, off"
               :: "v"(lds_off), "v"(gaddr)
               : "memory");
}
#define S_WAIT_ASYNCCNT(N) asm volatile("s_wait_asynccnt %0" :: "n"(N) : "memory")

// Stage one 513-float frame row into LDS. Called by wave 0 only.
// 16 fully-active b32 issues + 1 single-lane issue for column 512 => 17 async ops.
__device__ __forceinline__ void stage_row(uint32_t lds_base, const float* g, int lane) {
#pragma unroll
  for (int c = 0; c < 17; ++c) {
    int v = lane + c * 32;
    if (v < V_) async_ld_b32(lds_base + (uint32_t)v * 4u, g + v);
  }
}

__device__ __forceinline__ float wave_max(float v) {
#pragma unroll
  for (int off = 16; off > 0; off >>= 1) v = fmaxf(v, __shfl_xor(v, off, 32));
  return v;
}
__device__ __forceinline__ float wave_sum(float v) {
#pragma unroll
  for (int off = 16; off > 0; off >>= 1) v += __shfl_xor(v, off, 32);
  return v;
}

// ---------------- Phase 1: per-batch run-length segmentation ----------------
__global__ __launch_bounds__(BLK) void k_segmap(const int* __restrict__ align,
                                                const int* __restrict__ lengths,
                                                const int* __restrict__ blankp,
                                                int* __restrict__ segStart,
                                                int* __restrict__ segCount,
                                                float* __restrict__ outLen) {
  __shared__ int lab[T_];
  __shared__ int sStart[T_];
  __shared__ int sCount[T_];
  __shared__ int sN;
  const int b = blockIdx.x;
  const int tid = threadIdx.x;

  for (int t = tid; t < T_; t += BLK) lab[t] = align[(size_t)b * T_ + t];
  __syncthreads();

  if (tid == 0) {
    int len = lengths[b];
    if (len < 0) len = 0;
    if (len > T_) len = T_;
    const int blank = blankp[0];
    int n = 0, t = 0;
    while (t < len) {
      const int lbl = lab[t];
      const int s0 = t;
      while (t < len && lab[t] == lbl) ++t;
      if (lbl != blank) { sStart[n] = s0; sCount[n] = t - s0; ++n; }
    }
    sN = n;
    outLen[b] = (float)(n > 0 ? n : 1);  // empty utterance -> length 1
  }
  __syncthreads();

  const int n = sN;
  for (int d = tid; d < T_; d += BLK) {
    const size_t idx = (size_t)b * T_ + d;
    segStart[idx] = (d < n) ? sStart[d] : 0;
    segCount[idx] = (d < n) ? sCount[d] : 0;  // 0 => emit zero row
  }
}

// ------- Phase 2: fused softmax + segment mean-pool, async-LDS pipeline -----
// One block per packed row (b, d). Softmax stats (max, sum-exp) are computed
// from the LDS-staged row, so logits are read exactly once overall and invalid
// / blank frames are never read at all.
__global__ __launch_bounds__(BLK) void k_accum(const float* __restrict__ logits,
                                               const int* __restrict__ segStart,
                                               const int* __restrict__ segCount,
                                               float* __restrict__ packed) {
  const int d = blockIdx.x;   // dest (packed) row
  const int b = blockIdx.y;   // batch
  const int tid = threadIdx.x;
  float* out = packed + ((size_t)b * T_ + d) * V_;

  const int cnt = segCount[(size_t)b * T_ + d];
  if (cnt == 0) {  // unused row: write zeros (output buffer is poisoned)
    out[tid] = 0.0f;
    out[tid + 256] = 0.0f;
    if (tid == 0) out[512] = 0.0f;
    return;
  }

  const int start = segStart[(size_t)b * T_ + d];
  const float* rows = logits + ((size_t)b * T_ + start) * (size_t)V_;
  const float inv_cnt = 1.0f / (float)cnt;

  __shared__ __align__(16) float buf[2][520];  // double buffer, 513 used
  __shared__ float smax[8];                    // per-wave max partials
  __shared__ float ssum[8];                    // per-wave sum partials
  const int wave = tid >> 5;
  const int lane = tid & 31;
  const uint32_t lds0 = (uint32_t)(uintptr_t)(&buf[0][0]);
  const uint32_t lds1 = (uint32_t)(uintptr_t)(&buf[1][0]);

  if (wave == 0) stage_row(lds0, rows, lane);  // prologue: frame 0

  float acc0 = 0.0f, acc1 = 0.0f, acc2 = 0.0f;
  for (int i = 0; i < cnt; ++i) {
    if (wave == 0) {
      if (i + 1 < cnt) {
        // prefetch next frame into the other buffer, then wait for current
        stage_row((i & 1) ? lds0 : lds1, rows + (size_t)(i + 1) * V_, lane);
        S_WAIT_ASYNCCNT(17);   // 17 ops outstanding for "next"; current done
      } else {
        S_WAIT_ASYNCCNT(0);
      }
    }
    __syncthreads();  // A: buf[i&1] visible to all 8 waves

    const float* r = buf[i & 1];
    const float x0 = r[tid];
    const float x1 = r[tid + 256];
    const float x2 = (tid == 0) ? r[512] : -3.4e38f;

    // block max over 513 values
    float m = wave_max(fmaxf(fmaxf(x0, x1), x2));
    if (lane == 0) smax[wave] = m;
    __syncthreads();  // B
    float mx = smax[0];
#pragma unroll
    for (int wv = 1; wv < 8; ++wv) mx = fmaxf(mx, smax[wv]);

    // exponentials + block sum
    const float e0 = __expf(x0 - mx);
    const float e1 = __expf(x1 - mx);
    const float e2 = (tid == 0) ? __expf(x2 - mx) : 0.0f;
    float s = wave_sum(e0 + e1 + e2);
    if (lane == 0) ssum[wave] = s;
    __syncthreads();  // C
    float tot = ssum[0];
#pragma unroll
    for (int wv = 1; wv < 8; ++wv) tot += ssum[wv];

    const float w = __fdividef(inv_cnt, tot);  // (1/count) * (1/sum-exp)
    acc0 += e0 * w;
    acc1 += e1 * w;
    if (tid == 0) acc2 += e2 * w;
    // No extra barrier needed: buf[i&1] reads complete before barrier B, and
    // it is only restaged after barrier C of this iteration (next loop top).
  }

  out[tid] = acc0;
  out[tid + 256] = acc1;       // tid+256 <= 511 < 513
  if (tid == 0) out[512] = acc2;
}

// ---------------------------------------------------------------------------
extern "C" void kernel_launch(void* const* d_in, const int* in_sizes, int n_in,
                              void* d_out, int out_size, void* d_ws, size_t ws_size,
                              hipStream_t stream) {
  const float* logits = (const float*)d_in[0];   // [B, T, V] f32
  const int* align    = (const int*)d_in[1];     // [B, T] i32
  const int* lengths  = (const int*)d_in[2];     // [B] i32
  const int* blankp   = (const int*)d_in[3];     // [1] i32
  float* out = (float*)d_out;                    // [B,T,V] packed ++ [B] lengths

  // workspace: 2 * B*T * 4B = 512 KB
  int* segStart = (int*)d_ws;
  int* segCount = segStart + (size_t)B_ * T_;

  k_segmap<<<B_, BLK, 0, stream>>>(align, lengths, blankp, segStart, segCount,
                                   out + (size_t)B_ * T_ * V_);
  k_accum<<<dim3(T_, B_), BLK, 0, stream>>>(logits, segStart, segCount, out);

  (void)in_sizes; (void)n_in; (void)out_size; (void)ws_size;
}